// GraphAttentionModel_72619307040859
// MI455X (gfx1250) — compile-verified
//
#include <hip/hip_runtime.h>

typedef float v2f __attribute__((ext_vector_type(2)));
typedef float v8f __attribute__((ext_vector_type(8)));

#define NEG_SLOPE 0.2f

#if defined(__AMDGCN__) && __has_builtin(__builtin_amdgcn_global_load_async_to_lds_b32) && \
    __has_builtin(__builtin_amdgcn_s_wait_asynccnt)
#define USE_ASYNC_LDS 1
#else
#define USE_ASYNC_LDS 0
#endif

// ---- monotone float<->uint encoding for atomic segment-max -----------------
__device__ __forceinline__ unsigned enc_f32(float f) {
  unsigned u = __float_as_uint(f);
  return (u & 0x80000000u) ? ~u : (u | 0x80000000u);
}
__device__ __forceinline__ float dec_f32(unsigned u) {
  return __uint_as_float((u & 0x80000000u) ? (u & 0x7fffffffu) : ~u);
}
// enc(-inf) = ~0xFF800000 = 0x007FFFFF
#define ENC_NEG_INF 0x007FFFFFu

// ---- dense fc via fp32 WMMA: H[n x 64] = (relu?)X[n x in_f] @ W[in_f x 64] --
// One wave computes a 16x64 tile: 4 accumulators of 16x16, K-loop step 4.
// A layout (16x4 f32): lanes 0-15 -> M=lane, {K=k0,k0+1}; lanes 16-31 -> {K=k0+2,k0+3}.
// B layout (4x16 f32): symmetric, N = lane&15.
// LDS W is K-pair interleaved: Wswz[kp*128 + 2n + j] = W[2kp+j][n]
// so a lane's B operand {W[kk][n], W[kk+1][n]} is one contiguous b64.
__global__ __launch_bounds__(256) void fc_wmma_kernel(
    const float* __restrict__ X, const float* __restrict__ W,
    float* __restrict__ H, int n, int in_f, int relu_in)
{
  __shared__ float Wswz[(128 / 2) * 128];  // 32 KB

  const int tid = threadIdx.x;

#if USE_ASYNC_LDS
  // Async copy W -> LDS, swizzling via per-lane LDS destination addresses.
  for (int idx = tid; idx < in_f * 64; idx += 256) {
    const int k = idx >> 6;        // row of W
    const int nn = idx & 63;       // col of W
    const float* gsrc = W + idx;   // coalesced: consecutive lanes, consecutive dwords
    float* ldst = &Wswz[(k >> 1) * 128 + (nn << 1) + (k & 1)];
    __builtin_amdgcn_global_load_async_to_lds_b32(
        (__attribute__((address_space(1))) int*)gsrc,
        (__attribute__((address_space(3))) int*)ldst, 0, 0);
  }
  __builtin_amdgcn_s_wait_asynccnt(0);
#else
  for (int idx = tid; idx < (in_f >> 1) * 64; idx += 256) {
    const int kp = idx >> 6;
    const int nn = idx & 63;
    v2f w;
    w.x = W[(kp * 2) * 64 + nn];
    w.y = W[(kp * 2 + 1) * 64 + nn];
    *(v2f*)&Wswz[kp * 128 + (nn << 1)] = w;
  }
#endif
  __syncthreads();

  const int wave = tid >> 5;
  const int lane = tid & 31;
  const int row0 = (blockIdx.x * 8 + wave) * 16;
  if (row0 >= n) return;  // wave-uniform: EXEC stays all-1s for WMMA

  const int mcol = lane & 15;
  const int half = lane >> 4;

  v8f acc0 = {}, acc1 = {}, acc2 = {}, acc3 = {};

  int r = row0 + mcol;
  if (r >= n) r = n - 1;  // clamp loads; stores are guarded
  const float* __restrict__ xrow = X + (size_t)r * in_f + half * 2;

  // software-pipelined A fetch: prefetch next K-step before the 4 WMMAs
  v2f a_cur = *(const v2f*)(xrow);
  for (int k0 = 0; k0 < in_f; k0 += 4) {
    const int kn = (k0 + 4 < in_f) ? (k0 + 4) : 0;  // safe wrap on last iter
    v2f a_nxt = *(const v2f*)(xrow + kn);

    v2f a = a_cur;
    if (relu_in) { a.x = fmaxf(a.x, 0.f); a.y = fmaxf(a.y, 0.f); }

    const float* wb = &Wswz[(((k0 >> 1) + half) << 7) + (mcol << 1)];
    v2f b0 = *(const v2f*)(wb);
    v2f b1 = *(const v2f*)(wb + 32);
    v2f b2 = *(const v2f*)(wb + 64);
    v2f b3 = *(const v2f*)(wb + 96);

    acc0 = __builtin_amdgcn_wmma_f32_16x16x4_f32(false, a, false, b0, (short)0, acc0, false, false);
    acc1 = __builtin_amdgcn_wmma_f32_16x16x4_f32(false, a, false, b1, (short)0, acc1, false, false);
    acc2 = __builtin_amdgcn_wmma_f32_16x16x4_f32(false, a, false, b2, (short)0, acc2, false, false);
    acc3 = __builtin_amdgcn_wmma_f32_16x16x4_f32(false, a, false, b3, (short)0, acc3, false, false);

    a_cur = a_nxt;
  }

  // C/D layout: VGPR r8, lanes 0-15 -> M=r8, lanes 16-31 -> M=r8+8; N = lane&15.
  v8f accs[4] = {acc0, acc1, acc2, acc3};
  for (int t = 0; t < 4; ++t) {
    for (int r8 = 0; r8 < 8; ++r8) {
      int rr = row0 + r8 + 8 * half;
      if (rr < n) H[(size_t)rr * 64 + t * 16 + mcol] = accs[t][r8];
    }
  }
}

// ---- per-node attention dots: el = h . attn_l, er = h . attn_r --------------
__global__ __launch_bounds__(256) void node_attn_kernel(
    const float* __restrict__ H, const float* __restrict__ al,
    const float* __restrict__ ar, float* __restrict__ el,
    float* __restrict__ er, int n)
{
  int i = blockIdx.x * 256 + threadIdx.x;
  if (i >= n) return;
  const float4* hp = (const float4*)(H + (size_t)i * 64);
  const float4* ap = (const float4*)al;
  const float4* bp = (const float4*)ar;
  float sl = 0.f, sr = 0.f;
  for (int c = 0; c < 16; ++c) {
    float4 h4 = hp[c], a4 = ap[c], b4 = bp[c];
    sl += h4.x * a4.x + h4.y * a4.y + h4.z * a4.z + h4.w * a4.w;
    sr += h4.x * b4.x + h4.y * b4.y + h4.z * b4.z + h4.w * b4.w;
  }
  el[i] = sl;
  er[i] = sr;
}

// ---- seed aggregate buffer with bias, reset segment max/sum ----------------
__global__ __launch_bounds__(256) void init_node_kernel(
    float* __restrict__ agg, const float* __restrict__ bias,
    unsigned* __restrict__ mx, float* __restrict__ s, int n)
{
  int i = blockIdx.x * 256 + threadIdx.x;
  if (i >= n) return;
  mx[i] = ENC_NEG_INF;
  s[i] = 0.f;
  float4* ap = (float4*)(agg + (size_t)i * 64);
  const float4* bp = (const float4*)bias;
  for (int c = 0; c < 16; ++c) ap[c] = bp[c];
}

// ---- e = leaky(el[src]+er[dst]); segment max over dst ----------------------
__global__ __launch_bounds__(256) void edge_max_kernel(
    const int* __restrict__ src, const int* __restrict__ dst,
    const float* __restrict__ el, const float* __restrict__ er,
    float* __restrict__ ebuf, unsigned* __restrict__ mx, int E)
{
  int i = blockIdx.x * 256 + threadIdx.x;
  if (i >= E) return;
  float e = el[src[i]] + er[dst[i]];
  e = (e >= 0.f) ? e : NEG_SLOPE * e;
  ebuf[i] = e;
  atomicMax(&mx[dst[i]], enc_f32(e));
}

// ---- ex = exp(e - m[dst]); segment sum over dst ----------------------------
__global__ __launch_bounds__(256) void edge_sum_kernel(
    const int* __restrict__ dst, float* __restrict__ ebuf,
    const unsigned* __restrict__ mx, float* __restrict__ s, int E)
{
  int i = blockIdx.x * 256 + threadIdx.x;
  if (i >= E) return;
  int d = dst[i];
  float ex = expf(ebuf[i] - dec_f32(mx[d]));
  ebuf[i] = ex;
  atomicAdd(&s[d], ex);
}

// ---- agg[dst] += (ex/s[dst]) * h[src]; one thread per (edge, 4-col group) --
__global__ __launch_bounds__(256) void edge_agg_kernel(
    const int* __restrict__ src, const int* __restrict__ dst,
    const float* __restrict__ ebuf, const float* __restrict__ s,
    const float* __restrict__ H, float* __restrict__ agg, long total)
{
  long t = (long)blockIdx.x * 256 + threadIdx.x;
  if (t >= total) return;
  int e = (int)(t >> 4);
  int cg = ((int)t & 15) * 4;
  int d = dst[e];
  float a = ebuf[e] / s[d];
  float4 h4 = *(const float4*)(H + (size_t)src[e] * 64 + cg);
  float* ap = agg + (size_t)d * 64 + cg;
  atomicAdd(ap + 0, a * h4.x);
  atomicAdd(ap + 1, a * h4.y);
  atomicAdd(ap + 2, a * h4.z);
  atomicAdd(ap + 3, a * h4.w);
}

extern "C" void kernel_launch(void* const* d_in, const int* in_sizes, int n_in,
                              void* d_out, int out_size, void* d_ws, size_t ws_size,
                              hipStream_t stream)
{
  (void)n_in; (void)out_size; (void)ws_size;
  const float* feat = (const float*)d_in[0];
  // d_in[1] edge_weights: accepted but unused (matches reference)
  const int*   src  = (const int*)d_in[2];
  const int*   dst  = (const int*)d_in[3];
  const float* W1   = (const float*)d_in[4];
  const float* al1  = (const float*)d_in[5];
  const float* ar1  = (const float*)d_in[6];
  const float* b1   = (const float*)d_in[7];
  const float* W2   = (const float*)d_in[8];
  const float* al2  = (const float*)d_in[9];
  const float* ar2  = (const float*)d_in[10];
  const float* b2   = (const float*)d_in[11];

  const int n = in_sizes[0] / 128;   // features: [n, 128]
  const int E = in_sizes[2];         // src: [E]
  float* out = (float*)d_out;

  // workspace carve-up (~60 MB)
  float* p  = (float*)d_ws;
  float* h  = p; p += (size_t)n * 64;   // dense fc output (both layers)
  float* g  = p; p += (size_t)n * 64;   // layer-1 aggregate (input to layer 2)
  float* el = p; p += n;
  float* er = p; p += n;
  unsigned* mx = (unsigned*)p; p += n;
  float* s  = p; p += n;
  float* eb = p; p += E;

  const int TB = 256;
  const int nodeBlocks = (n + TB - 1) / TB;
  const int edgeBlocks = (E + TB - 1) / TB;
  const long aggTotal  = (long)E * 16;
  const int aggBlocks  = (int)((aggTotal + TB - 1) / TB);
  const int fcBlocks   = (((n + 15) / 16) + 7) / 8;  // 8 waves x 16 rows per block

  // ---------------- layer 1 ----------------
  fc_wmma_kernel<<<fcBlocks, TB, 0, stream>>>(feat, W1, h, n, 128, 0);
  node_attn_kernel<<<nodeBlocks, TB, 0, stream>>>(h, al1, ar1, el, er, n);
  init_node_kernel<<<nodeBlocks, TB, 0, stream>>>(g, b1, mx, s, n);
  edge_max_kernel<<<edgeBlocks, TB, 0, stream>>>(src, dst, el, er, eb, mx, E);
  edge_sum_kernel<<<edgeBlocks, TB, 0, stream>>>(dst, eb, mx, s, E);
  edge_agg_kernel<<<aggBlocks, TB, 0, stream>>>(src, dst, eb, s, h, g, aggTotal);

  // ---------------- layer 2 (relu fused into fc A-load) ----------------
  fc_wmma_kernel<<<fcBlocks, TB, 0, stream>>>(g, W2, h, n, 64, 1);
  node_attn_kernel<<<nodeBlocks, TB, 0, stream>>>(h, al2, ar2, el, er, n);
  init_node_kernel<<<nodeBlocks, TB, 0, stream>>>(out, b2, mx, s, n);
  edge_max_kernel<<<edgeBlocks, TB, 0, stream>>>(src, dst, el, er, eb, mx, E);
  edge_sum_kernel<<<edgeBlocks, TB, 0, stream>>>(dst, eb, mx, s, E);
  edge_agg_kernel<<<aggBlocks, TB, 0, stream>>>(src, dst, eb, s, h, out, aggTotal);
}